// GsumLayer_19172734010021
// MI455X (gfx1250) — compile-verified
//
#include <hip/hip_runtime.h>

typedef __attribute__((ext_vector_type(2))) float v2f;
typedef __attribute__((ext_vector_type(8))) float v8f;

// Problem shape (reference: B=8, N=4096, D=32, fp32)
constexpr int BATCH = 8;
constexpr int N     = 4096;
constexpr int D     = 32;

constexpr int KC          = 128;      // k-chunk staged in LDS per block
constexpr int LDS_STRIDE  = KC + 2;   // even stride -> keeps float2 (8B) alignment in LDS
constexpr int WAVES       = 8;        // waves per block (wave32)
constexpr int THREADS     = WAVES * 32;
constexpr int ROWS_PER_BLOCK = WAVES * 16;   // one 16-row tile per wave

__global__ __launch_bounds__(THREADS)
void gsum_wmma_f32_kernel(const float* __restrict__ x,   // [B, N, D]
                          const float* __restrict__ a,   // [B, N, N]
                          float* __restrict__ out) {     // [B, N, D]
  __shared__ float xT[D * LDS_STRIDE];   // transposed x chunk: xT[d][k - k0]

  const int tid  = threadIdx.x;
  const int wave = tid >> 5;
  const int lane = tid & 31;
  const int g    = lane >> 4;   // lane group: 0 -> lanes 0..15, 1 -> lanes 16..31
  const int hn   = lane & 15;   // M index (for A) / N index (for B, C, D)

  const int blocksPerBatch = N / ROWS_PER_BLOCK;          // 32
  const int b    = blockIdx.x / blocksPerBatch;
  const int rb   = blockIdx.x % blocksPerBatch;
  const int row0 = rb * ROWS_PER_BLOCK + wave * 16;       // this wave's 16-row tile

  const float* xb      = x + (size_t)b * N * D;
  // A operand layout (V_WMMA_F32_16X16X4_F32, 16x4 fp32):
  //   lanes 0..15  hold (M=lane,   K=k+0..k+1) in v[0..1]
  //   lanes 16..31 hold (M=lane-16,K=k+2..k+3) in v[0..1]
  const float* aRowBase = a + ((size_t)b * N + row0 + hn) * N;  // this lane's row of a
  const float* aRow     = aRowBase + 2 * g;                     // + lane-group K offset

  v8f acc0 = {};   // out cols  0..15 of the 16-row tile
  v8f acc1 = {};   // out cols 16..31

  for (int k0 = 0; k0 < N; k0 += KC) {
    // ---- cooperative load of x[b][k0:k0+KC][:] transposed into LDS ----
    {
      const float4* src = (const float4*)(xb + (size_t)k0 * D);
      #pragma unroll
      for (int j = 0; j < (KC * D) / (4 * THREADS); ++j) {   // 4 float4 per thread
        const int f  = tid + j * THREADS;   // float4 index within the chunk
        float4 v     = src[f];              // coalesced global_load_b128
        const int kk = f >> 3;              // k within chunk (D/4 = 8 float4 per k-row)
        const int d0 = (f & 7) * 4;
        xT[(d0 + 0) * LDS_STRIDE + kk] = v.x;
        xT[(d0 + 1) * LDS_STRIDE + kk] = v.y;
        xT[(d0 + 2) * LDS_STRIDE + kk] = v.z;
        xT[(d0 + 3) * LDS_STRIDE + kk] = v.w;
      }
    }
    __syncthreads();

    // ---- prefetch next A chunk: 2 lines/lane * 2 lane groups = full 512B/row span ----
    if (k0 + KC < N) {
      const char* pf = (const char*)(aRowBase + k0 + KC);
      __builtin_prefetch(pf + g * 256, 0, 0);        // global_prefetch_b8
      __builtin_prefetch(pf + 128 + g * 256, 0, 0);
    }

    // ---- WMMA over this k-chunk: 32 steps of K=4 ----
    const float* pA  = aRow + k0;                              // global, 8B aligned
    const float* pB0 = &xT[(hn +  0) * LDS_STRIDE + 2 * g];    // LDS,    8B aligned
    const float* pB1 = &xT[(hn + 16) * LDS_STRIDE + 2 * g];

    #pragma unroll 8
    for (int kk = 0; kk < KC; kk += 4) {
      v2f av = *(const v2f*)(pA  + kk);   // global_load_b64: a[row][k0+kk+2g .. +1]
      v2f b0 = *(const v2f*)(pB0 + kk);   // ds_load_b64:     xT[n     ][kk+2g .. +1]
      v2f b1 = *(const v2f*)(pB1 + kk);   // ds_load_b64:     xT[n + 16][kk+2g .. +1]

      // D = A(16x4) x B(4x16) + C ; 8 args: (neg_a, A, neg_b, B, c_mod, C, reuse_a, reuse_b)
      acc0 = __builtin_amdgcn_wmma_f32_16x16x4_f32(false, av, false, b0,
                                                   (short)0, acc0, false, false);
      acc1 = __builtin_amdgcn_wmma_f32_16x16x4_f32(false, av, false, b1,
                                                   (short)0, acc1, false, false);
    }
    __syncthreads();   // xT reused next chunk
  }

  // ---- store C/D tiles: VGPR r -> (M = r + 8*g, N = hn) ----
  float* orow = out + ((size_t)b * N + row0) * D;
  #pragma unroll
  for (int r = 0; r < 8; ++r) {
    const int m = r + 8 * g;
    orow[(size_t)m * D + hn]      = acc0[r];
    orow[(size_t)m * D + 16 + hn] = acc1[r];
  }
}

extern "C" void kernel_launch(void* const* d_in, const int* in_sizes, int n_in,
                              void* d_out, int out_size, void* d_ws, size_t ws_size,
                              hipStream_t stream) {
  const float* x = (const float*)d_in[0];   // [8, 4096, 32]
  const float* a = (const float*)d_in[1];   // [8, 4096, 4096]
  float* out     = (float*)d_out;           // [8, 4096, 32]
  (void)in_sizes; (void)n_in; (void)out_size; (void)d_ws; (void)ws_size;

  const int grid = BATCH * (N / ROWS_PER_BLOCK);   // 8 * 32 = 256 blocks, 2048 waves
  gsum_wmma_f32_kernel<<<grid, THREADS, 0, stream>>>(x, a, out);
}